// FreeConvNetwork_33767032882016
// MI455X (gfx1250) — compile-verified
//
#include <hip/hip_runtime.h>

typedef __attribute__((ext_vector_type(2))) float v2f;
typedef __attribute__((ext_vector_type(8))) float v8f;

#define BATCH 16

// ---------------------------------------------------------------------------
// Broadcast-init: dst[i] = src[i % period]  (writes bias into accumulators)
// ---------------------------------------------------------------------------
__global__ void bcast_init(float* __restrict__ dst, const float* __restrict__ src,
                           unsigned total, unsigned period) {
    unsigned i = blockIdx.x * blockDim.x + threadIdx.x;
    if (i < total) dst[i] = src[i % period];
}

// ---------------------------------------------------------------------------
// ReLU + 2x2 maxpool:  acc (16,8,213,173) -> out (16,8,106,86)
// relu(max4) == max(relu of each), relu is monotonic.
// ---------------------------------------------------------------------------
__global__ void relu_maxpool2(const float* __restrict__ acc, float* __restrict__ out) {
    const int H = 213, W = 173, PH = 106, PW = 86;
    unsigned i = blockIdx.x * blockDim.x + threadIdx.x;
    unsigned total = (unsigned)BATCH * 8 * PH * PW;
    if (i >= total) return;
    int pw = (int)(i % PW); unsigned t = i / PW;
    int ph = (int)(t % PH); t /= PH;            // t = b*8 + o
    const float* p = acc + ((long)t * H + 2 * ph) * W + 2 * pw;
    float v = fmaxf(fmaxf(p[0], p[1]), fmaxf(p[W], p[W + 1]));
    out[i] = fmaxf(v, 0.f);
}

// ---------------------------------------------------------------------------
// Locally-connected conv via V_WMMA_F32_16X16X4_F32.
// One wave = one (h,w) position x one 16-wide out-channel tile x one channel
// chunk.  GEMM per position: [16 batch x C*36] @ [C*36 x O].
//   A (16x4): lane -> batch m = lane&15, K pair = kbase + 2*half + {0,1}
//   B (4x16): lane -> out-ch  n = lane&15, same K pair
//   C (16x16): VGPR v, lane -> M = v + 8*half, N = lane&15
// K order = (channel, tap 0..35).  Tap loop fully unrolled: tap offsets are
// compile-time constants selected by `half`; since k-pairs start at even taps,
// tap%6 <= 4, so (a0,a1) are always row-contiguous -> single b64 load.
// Out-of-range out-channel lanes load a clamped valid address (no EXEC churn
// in the loop); their D columns are discarded at the masked store.
// Partial sums atomically added into bias-pre-initialized acc (16,O,H,W).
// ---------------------------------------------------------------------------
template <int C, int O, int H, int W, int INH, int INW, int STRIDE,
          int NT, int KSPLIT, bool RELU_IN>
__global__ __launch_bounds__(256)
void lc2d_wmma(const float* __restrict__ in,   // (16, C, INH, INW)
               const float* __restrict__ wgt,  // (O, C, H, W, 36)
               float* __restrict__ acc) {      // (16, O, H, W)
    static_assert(C % KSPLIT == 0, "channel chunking");
    const int CPER = C / KSPLIT;
    const int lane = threadIdx.x & 31;
    int wid = blockIdx.x * (blockDim.x >> 5) + (threadIdx.x >> 5);
    const int total = H * W * NT * KSPLIT;
    if (wid >= total) return;                  // wave-uniform exit (EXEC stays full)

    int kc = wid % KSPLIT; wid /= KSPLIT;
    int nt = wid % NT;     wid /= NT;
    const int h = wid / W, w = wid % W;

    const int m    = lane & 15;                // batch row of A
    const int half = lane >> 4;                // K-pair select
    const int oc   = nt * 16 + (lane & 15);    // out channel (B column)
    const int occ  = oc < O ? oc : O - 1;      // clamped: loads always legal

    v8f c = {0.f, 0.f, 0.f, 0.f, 0.f, 0.f, 0.f, 0.f};

    const int c0 = kc * CPER;
    for (int cc = c0; cc < c0 + CPER; ++cc) {
        // Per-lane base pointers for this channel
        const float* pa  = in + (((long)m * C + cc) * INH + h * STRIDE) * INW
                              + w * STRIDE;
        const float* pbh = wgt + ((((long)occ * C + cc) * H + h) * W + w) * 36
                               + 2 * half;
#pragma unroll
        for (int tb = 0; tb < 36; tb += 4) {
            // Compile-time tap->pixel offsets for half=0 (tap tb) / half=1 (tap tb+2)
            const int T2   = tb + 2;
            const int OFF0 = (tb / 6) * INW + (tb % 6);
            const int OFF2 = (T2 / 6) * INW + (T2 % 6);
            const int o0   = half ? OFF2 : OFF0;   // one cndmask

            float a0 = pa[o0];
            float a1 = pa[o0 + 1];                 // same 6-wide row: contiguous
            if (RELU_IN) { a0 = fmaxf(a0, 0.f); a1 = fmaxf(a1, 0.f); }

            float b0 = __builtin_nontemporal_load(pbh + tb);      // imm offset
            float b1 = __builtin_nontemporal_load(pbh + tb + 1);  // merges to b64

            v2f A  = {a0, a1};
            v2f Bv = {b0, b1};
            c = __builtin_amdgcn_wmma_f32_16x16x4_f32(
                    /*neg_a=*/false, A, /*neg_b=*/false, Bv,
                    /*c_mod=*/(short)0, c, /*reuse_a=*/false, /*reuse_b=*/false);
        }
    }

    if (oc < O) {
#pragma unroll
        for (int v = 0; v < 8; ++v) {
            int mb = v + 8 * half;
            atomicAdd(&acc[(((long)mb * O + oc) * H + h) * W + w], c[v]);
        }
    }
}

// ---------------------------------------------------------------------------
// FC: out(16,10178) += relu(h)(16,6144) @ wf(10178,6144)^T
// One wave = one 16-wide N tile x one of 16 K chunks (10192 waves to keep
// the 250 MB wf stream at full HBM bandwidth).  Out-of-range N lanes load a
// clamped valid row; their columns are discarded at the masked store.
// ---------------------------------------------------------------------------
__global__ __launch_bounds__(256)
void fc_wmma(const float* __restrict__ hbuf,  // (16, 6144), relu applied on read
             const float* __restrict__ wf,    // (10178, 6144)
             float* __restrict__ out) {       // (16, 10178), bias pre-initialized
    const int KTOT = 6144, N = 10178, NTILES = 637, KSPLIT = 16;
    const int KPER = KTOT / KSPLIT;            // 384 -> 96 WMMA steps
    const int lane = threadIdx.x & 31;
    int wid = blockIdx.x * (blockDim.x >> 5) + (threadIdx.x >> 5);
    if (wid >= NTILES * KSPLIT) return;

    int kc = wid % KSPLIT;
    int nt = wid / KSPLIT;

    const int m    = lane & 15;
    const int half = lane >> 4;
    const int nc   = nt * 16 + (lane & 15);
    const int ncc  = nc < N ? nc : N - 1;      // clamped: loads always legal

    const int k0 = kc * KPER;
    const float* pa = hbuf + (long)m * KTOT + k0 + 2 * half;
    const float* pb = wf + (long)ncc * KTOT + k0 + 2 * half;

    v8f c = {0.f, 0.f, 0.f, 0.f, 0.f, 0.f, 0.f, 0.f};

    for (int kb = 0; kb < KPER; kb += 4) {
        // A: activations, heavily reused (0.4 MB, L2-resident) -> cached loads
        float a0 = fmaxf(pa[kb],     0.f);
        float a1 = fmaxf(pa[kb + 1], 0.f);
        // B: wf rows are B columns; streamed once -> non-temporal b64
        float b0 = __builtin_nontemporal_load(pb + kb);
        float b1 = __builtin_nontemporal_load(pb + kb + 1);

        v2f A  = {a0, a1};
        v2f Bv = {b0, b1};
        c = __builtin_amdgcn_wmma_f32_16x16x4_f32(
                false, A, false, Bv, (short)0, c, false, false);
    }

    if (nc < N) {
#pragma unroll
        for (int v = 0; v < 8; ++v) {
            int mb = v + 8 * half;
            atomicAdd(&out[(long)mb * N + nc], c[v]);
        }
    }
}

// ---------------------------------------------------------------------------
extern "C" void kernel_launch(void* const* d_in, const int* in_sizes, int n_in,
                              void* d_out, int out_size, void* d_ws, size_t ws_size,
                              hipStream_t stream) {
    (void)in_sizes; (void)n_in; (void)out_size; (void)ws_size;
    const float* x  = (const float*)d_in[0];   // (16,3,218,178)
    const float* w1 = (const float*)d_in[1];   // (8,3,213,173,36)
    const float* b1 = (const float*)d_in[2];   // (8,213,173)
    const float* w2 = (const float*)d_in[3];   // (16,8,51,41,36)
    const float* b2 = (const float*)d_in[4];   // (16,51,41)
    const float* w3 = (const float*)d_in[5];   // (32,16,16,12,36)
    const float* b3 = (const float*)d_in[6];   // (32,16,12)
    const float* wf = (const float*)d_in[7];   // (10178,6144)
    const float* bf = (const float*)d_in[8];   // (10178)
    float* out = (float*)d_out;                // (16,10178)

    // Workspace layout (floats): ~26 MB total
    float* acc1   = (float*)d_ws;              // 16*8*213*173 = 4,716,672
    float* pooled = acc1 + 4716672;            // 16*8*106*86  = 1,166,848
    float* acc2   = pooled + 1166848;          // 16*16*51*41  =   535,296
    float* acc3   = acc2 + 535296;             // 16*32*16*12  =    98,304

    // Bias pre-init of accumulators (and d_out, which the harness poisons)
    { unsigned n = 4716672; bcast_init<<<(n + 255) / 256, 256, 0, stream>>>(acc1, b1, n, 8 * 213 * 173); }
    { unsigned n = 535296;  bcast_init<<<(n + 255) / 256, 256, 0, stream>>>(acc2, b2, n, 16 * 51 * 41); }
    { unsigned n = 98304;   bcast_init<<<(n + 255) / 256, 256, 0, stream>>>(acc3, b3, n, 32 * 16 * 12); }
    { unsigned n = BATCH * 10178;
      bcast_init<<<(n + 255) / 256, 256, 0, stream>>>(out, bf, n, 10178); }

    // Layer 1: C=3,O=8, 213x173, stride 1, K=108, 36,849 waves
    {
        const int waves = 213 * 173;
        lc2d_wmma<3, 8, 213, 173, 218, 178, 1, 1, 1, false>
            <<<(waves + 7) / 8, 256, 0, stream>>>(x, w1, acc1);
    }
    // ReLU + maxpool
    {
        unsigned n = BATCH * 8 * 106 * 86;
        relu_maxpool2<<<(n + 255) / 256, 256, 0, stream>>>(acc1, pooled);
    }
    // Layer 2: C=8,O=16, 51x41, stride 2, K=288 split 2 -> 4,182 waves
    {
        const int waves = 51 * 41 * 2;
        lc2d_wmma<8, 16, 51, 41, 106, 86, 2, 1, 2, false>
            <<<(waves + 7) / 8, 256, 0, stream>>>(pooled, w2, acc2);
    }
    // Layer 3: C=16,O=32, 16x12, stride 3, K=576 split 4, 2 N-tiles -> 1,536 waves
    {
        const int waves = 16 * 12 * 2 * 4;
        lc2d_wmma<16, 32, 16, 12, 51, 41, 3, 2, 4, true>
            <<<(waves + 7) / 8, 256, 0, stream>>>(acc2, w3, acc3);
    }
    // FC: M=16, K=6144 split 16, N=10178 -> 10,192 waves
    {
        const int waves = 637 * 16;
        fc_wmma<<<(waves + 7) / 8, 256, 0, stream>>>(acc3, wf, out);
    }
}